// ResidualAverageBlock_41747082117850
// MI455X (gfx1250) — compile-verified
//
#include <hip/hip_runtime.h>

typedef __attribute__((ext_vector_type(2))) float v2f;
typedef __attribute__((ext_vector_type(8))) float v8f;

// ---------------------------------------------------------------------------
// Kernel 1: weighted reduction  s[c] = sum_n w[n]*x[n][c],  sw = sum_n w[n]
// float4-vectorized: each thread owns a fixed 4-column group (tid&7), each
// wave iteration covers 4 rows via b128 loads (512B per wave vmem issue).
// w[row] is uniform per 8-lane group; lane with (tid&7)==0 owns the row's
// sw contribution, so every row's weight is counted exactly once.
// Per-block partials written to workspace (deterministic tree, no atomics).
// ---------------------------------------------------------------------------
__global__ void reduce_kernel(const float* __restrict__ x,
                              const float* __restrict__ w,
                              float* __restrict__ s_part,   // [nb][32]
                              float* __restrict__ sw_part,  // [nb]
                              long long nVec4)              // N*8
{
    __shared__ float4 lds_s[256];
    __shared__ float  lds_sw[32];

    const int tid = threadIdx.x;
    const int cg  = tid & 7;                    // column group: cols 4*cg..4*cg+3
    const long long stride = (long long)gridDim.x * blockDim.x;

    float4 acc = {0.f, 0.f, 0.f, 0.f};
    float  accw = 0.0f;
    const float4* __restrict__ x4 = (const float4*)x;
    for (long long g = (long long)blockIdx.x * blockDim.x + tid; g < nVec4; g += stride) {
        const long long row = g >> 3;           // 8 float4 per 32-float row
        const float wv = w[row];                // uniform per 8-lane group
        const float4 v = x4[g];
        acc.x += wv * v.x;
        acc.y += wv * v.y;
        acc.z += wv * v.z;
        acc.w += wv * v.w;
        if (cg == 0) accw += wv;                // count each row's w exactly once
    }

    lds_s[tid] = acc;
    if (cg == 0) lds_sw[tid >> 3] = accw;
    __syncthreads();

    if (tid < 32) {                             // fold column c = tid
        const int grp = tid >> 2;               // which float4 group holds col c
        const int cmp = tid & 3;                // which component
        float sum = 0.0f;
        #pragma unroll
        for (int k = 0; k < 32; ++k) {
            const float4 v = lds_s[grp + 8 * k];
            sum += (cmp == 0) ? v.x : (cmp == 1) ? v.y : (cmp == 2) ? v.z : v.w;
        }
        s_part[(long long)blockIdx.x * 32 + tid] = sum;
    }
    if (tid == 0) {
        float sum = 0.0f;
        #pragma unroll
        for (int i = 0; i < 32; ++i) sum += lds_sw[i];
        sw_part[blockIdx.x] = sum;
    }
}

// ---------------------------------------------------------------------------
// Kernel 2: single wave.  Fold per-block partials, then
//   t[128]  = relu( (W_in @ s) / sw )     via V_WMMA_F32_16X16X4_F32
//   xadd[32]=  W_out @ t                  via V_WMMA_F32_16X16X4_F32
// A-matrix 16x4 fp32 layout (ISA 7.12.2): lane L holds M=L%16,
//   VGPR0=K0+2*(L/16), VGPR1=K0+2*(L/16)+1.  B (4x16) broadcast across N.
// D layout: acc[r] = D[M = r + 8*(L/16)][N = L%16]; all N columns equal.
// ---------------------------------------------------------------------------
__global__ void middle_kernel(const float* __restrict__ s_part,
                              const float* __restrict__ sw_part,
                              const float* __restrict__ W_in,   // [128][32]
                              const float* __restrict__ W_out,  // [32][128]
                              float* __restrict__ xadd,         // [32]
                              int nb)
{
    __shared__ float sv[32];
    __shared__ float tv[128];

    const int lane = threadIdx.x;      // 32 threads
    const int half = lane >> 4;        // 0 | 1
    const int mrow = lane & 15;

    float s = 0.0f, sw = 0.0f;
    for (int b = 0; b < nb; ++b) {
        s  += s_part[(long long)b * 32 + lane];
        sw += sw_part[b];
    }
    sv[lane] = s;
    __syncthreads();
    const float inv_sw = 1.0f / sw;    // uniform across lanes

    // ---- t = relu(W_in @ s * inv_sw) : 8 M-tiles x 8 K-tiles ----
    #pragma unroll
    for (int mt = 0; mt < 8; ++mt) {
        v8f acc = {};
        #pragma unroll
        for (int kt = 0; kt < 8; ++kt) {
            const int k0 = kt * 4 + half * 2;
            v2f a, b;
            a.x = W_in[(mt * 16 + mrow) * 32 + k0];
            a.y = W_in[(mt * 16 + mrow) * 32 + k0 + 1];
            b.x = sv[k0];
            b.y = sv[k0 + 1];
            acc = __builtin_amdgcn_wmma_f32_16x16x4_f32(
                false, a, false, b, (short)0, acc, false, false);
        }
        if (mrow == 0) {               // lanes 0 & 16 cover rows r and r+8
            #pragma unroll
            for (int r = 0; r < 8; ++r) {
                const float v = acc[r] * inv_sw;
                tv[mt * 16 + r + 8 * half] = v > 0.0f ? v : 0.0f;
            }
        }
    }
    __syncthreads();

    // ---- xadd = W_out @ t : 2 M-tiles x 32 K-tiles ----
    #pragma unroll
    for (int mt = 0; mt < 2; ++mt) {
        v8f acc = {};
        #pragma unroll
        for (int kt = 0; kt < 32; ++kt) {
            const int k0 = kt * 4 + half * 2;
            v2f a, b;
            a.x = W_out[(mt * 16 + mrow) * 128 + k0];
            a.y = W_out[(mt * 16 + mrow) * 128 + k0 + 1];
            b.x = tv[k0];
            b.y = tv[k0 + 1];
            acc = __builtin_amdgcn_wmma_f32_16x16x4_f32(
                false, a, false, b, (short)0, acc, false, false);
        }
        if (mrow == 0) {
            #pragma unroll
            for (int r = 0; r < 8; ++r) xadd[mt * 16 + r + 8 * half] = acc[r];
        }
    }
}

// ---------------------------------------------------------------------------
// Kernel 3: out[n][:] = x[n][:] + xadd[:]  (float4-vectorized, 2nd read of x
// should largely hit the 192MB L2)
// ---------------------------------------------------------------------------
__global__ void add_kernel(const float* __restrict__ x,
                           const float* __restrict__ xadd,  // [32], 16B aligned
                           float* __restrict__ out,
                           long long nVec4)                 // N*8
{
    __shared__ float4 xa[8];
    if (threadIdx.x < 8) xa[threadIdx.x] = ((const float4*)xadd)[threadIdx.x];
    __syncthreads();

    const long long stride = (long long)gridDim.x * blockDim.x;
    for (long long g = (long long)blockIdx.x * blockDim.x + threadIdx.x;
         g < nVec4; g += stride) {
        float4 v = ((const float4*)x)[g];
        const float4 a = xa[g & 7];            // (4g mod 32)/4
        v.x += a.x; v.y += a.y; v.z += a.z; v.w += a.w;
        ((float4*)out)[g] = v;
    }
}

extern "C" void kernel_launch(void* const* d_in, const int* in_sizes, int n_in,
                              void* d_out, int out_size, void* d_ws, size_t ws_size,
                              hipStream_t stream)
{
    const float* x     = (const float*)d_in[0];   // [N,32]
    const float* w     = (const float*)d_in[1];   // [N,1]
    const float* W_in  = (const float*)d_in[2];   // [128,32]
    const float* W_out = (const float*)d_in[3];   // [32,128]
    float* out = (float*)d_out;

    const long long N     = in_sizes[1];
    const long long nVec4 = N * 8;

    // workspace layout: [nb*32] s partials | [nb] sw partials | [32] xadd
    int nb = 1024;
    {
        const long long cap = ((long long)ws_size / 4 - 32) / 33;
        if (cap < nb) nb = (cap > 3) ? (int)(cap & ~3LL) : 1;
    }
    float* s_part  = (float*)d_ws;
    float* sw_part = s_part + (long long)nb * 32;
    float* xadd    = s_part + (long long)nb * 33;   // 16B-aligned for nb%4==0

    reduce_kernel<<<nb, 256, 0, stream>>>(x, w, s_part, sw_part, nVec4);
    middle_kernel<<<1, 32, 0, stream>>>(s_part, sw_part, W_in, W_out, xadd, nb);

    long long blocks64 = (nVec4 + 255) / 256;
    int blocks = (blocks64 > 2048) ? 2048 : (int)blocks64;
    add_kernel<<<blocks, 256, 0, stream>>>(x, xadd, out, nVec4);
}